// HeteroGCN_71683004170372
// MI455X (gfx1250) — compile-verified
//
#include <hip/hip_runtime.h>
#include <hip/hip_bf16.h>

// ---------------------------------------------------------------------------
// HeteroGCN forward for MI455X (gfx1250, wave32).
// GEMMs on V_WMMA_F32_16X16X4_F32 (fp32-exact matrix pipe); A tiles staged in
// LDS with conflict-free padded strides; edge phase (the roofline bottleneck,
// ~1.5-2 GB of gather + f32 atomics @ 23.3 TB/s) uses wave-per-edge float4
// gathers with global_prefetch_b8 lookahead.
// ---------------------------------------------------------------------------

#define NA 131072
#define NW 32768
#define DD 128
#define HH 128
#define E1 1048576
#define E2 262144
#define LH 256
#define OUTC 10

// padded LDS row strides (floats): keep 16B alignment for float4 stores,
// 8B alignment for b64 reads, and spread 16 rows over distinct banks.
#define AS_STRIDE 132   // for 128-wide tiles   (132 % 64 == 4)
#define X1_STRIDE 260   // for 256-wide x1 tile (260 % 64 == 4)
#define X2_STRIDE 36    // for 32-wide x2 tile  (36 % 64 == 36)

typedef float v2f __attribute__((ext_vector_type(2)));
typedef float v8f __attribute__((ext_vector_type(8)));

// -------------------------------------------------------------------------
// Edge scatter: one wave (32 lanes) per edge; each lane handles 4 dims.
// msg = feat[src] * w ; atomic-add into agg[dst], count into cnt[dst].
// -------------------------------------------------------------------------
__global__ __launch_bounds__(256) void scatter_edges(
    const float* __restrict__ feat, const int* __restrict__ src,
    const int* __restrict__ dst, const float* __restrict__ w,
    float* __restrict__ agg, float* __restrict__ cnt, int nedges) {
  int tid = blockIdx.x * blockDim.x + threadIdx.x;
  int e = tid >> 5;
  int lane = tid & 31;
  if (e >= nedges) return;
  // prefetch the gather row of an edge ~64 ahead (global_prefetch_b8)
  int ep = e + 64;
  if (ep < nedges) {
    int sp = src[ep];
    __builtin_prefetch(feat + (size_t)sp * DD + lane * 4, 0, 1);
  }
  int s = src[e];
  int d = dst[e];
  float wv = w[e];
  const float4* fs = (const float4*)(feat + (size_t)s * DD);
  float4 v = fs[lane];
  float* ap = agg + (size_t)d * DD + lane * 4;
  atomicAdd(ap + 0, v.x * wv);
  atomicAdd(ap + 1, v.y * wv);
  atomicAdd(ap + 2, v.z * wv);
  atomicAdd(ap + 3, v.w * wv);
  if (lane == 0) atomicAdd(cnt + d, 1.0f);
}

// mean: agg[i,d] /= max(cnt[i],1)
__global__ __launch_bounds__(256) void seg_div(
    float* __restrict__ agg, const float* __restrict__ cnt, int ntot) {
  int idx = blockIdx.x * blockDim.x + threadIdx.x;
  if (idx >= ntot) return;
  float c = cnt[idx >> 7];  // D == 128
  agg[idx] = agg[idx] / fmaxf(c, 1.0f);
}

// -------------------------------------------------------------------------
// Dual GEMM: C[M x 128] = A1@B1 + A2@B2 + bias, K = 128.
// One block = 8 waves = 16 output rows; wave wv owns 16x16 tile at n0=wv*16.
// Both 16x128 A tiles are staged in LDS once per block (8x fewer A reads).
// WMMA f32 layouts: A lane-half selects K pair {0,1}/{2,3}; B mirrors A;
// C/D: vgpr v -> row v + 8*(lane>=16), col = lane&15.
// -------------------------------------------------------------------------
__global__ __launch_bounds__(256) void gemm_dual_128(
    const float* __restrict__ A1, const float* __restrict__ B1,
    const float* __restrict__ A2, const float* __restrict__ B2,
    const float* __restrict__ bias, float* __restrict__ C) {
  __shared__ __align__(16) float As1[16 * AS_STRIDE];
  __shared__ __align__(16) float As2[16 * AS_STRIDE];
  const int lane = threadIdx.x & 31;
  const int wv = threadIdx.x >> 5;
  const int m0 = blockIdx.x * 16;
  const int l15 = lane & 15;
  const int koff = (lane >> 4) * 2;
  const int rhalf = (lane >> 4) * 8;
  const int n0 = wv * 16;

  // cooperative stage: 16 rows x 128 cols = 512 float4 per matrix
  for (int i = threadIdx.x; i < 512; i += 256) {
    int r = i >> 5;
    int c = (i & 31) << 2;
    float4 v1 = *(const float4*)(A1 + (size_t)(m0 + r) * DD + c);
    float4 v2 = *(const float4*)(A2 + (size_t)(m0 + r) * DD + c);
    *(float4*)(As1 + r * AS_STRIDE + c) = v1;
    *(float4*)(As2 + r * AS_STRIDE + c) = v2;
  }
  __syncthreads();

  const float* a1p = As1 + l15 * AS_STRIDE + koff;
  const float* a2p = As2 + l15 * AS_STRIDE + koff;
  v8f acc = {};
  for (int k = 0; k < DD; k += 4) {
    v2f a, b;
    a.x = a1p[k];
    a.y = a1p[k + 1];
    const float* bp = B1 + (size_t)(k + koff) * HH + n0 + l15;
    b.x = bp[0];
    b.y = bp[HH];
    acc = __builtin_amdgcn_wmma_f32_16x16x4_f32(false, a, false, b,
                                                (short)0, acc, false, false);
    a.x = a2p[k];
    a.y = a2p[k + 1];
    bp = B2 + (size_t)(k + koff) * HH + n0 + l15;
    b.x = bp[0];
    b.y = bp[HH];
    acc = __builtin_amdgcn_wmma_f32_16x16x4_f32(false, a, false, b,
                                                (short)0, acc, false, false);
  }
  float bb = bias[n0 + l15];
#pragma unroll
  for (int v = 0; v < 8; ++v)
    C[(size_t)(m0 + rhalf + v) * HH + n0 + l15] = acc[v] + bb;
}

// h_word = (h_rel + feat_w) * 0.5   (H == D so h_id == feat_w)
__global__ __launch_bounds__(256) void word_combine(
    float* __restrict__ hw, const float* __restrict__ fw, int ntot) {
  int idx = blockIdx.x * blockDim.x + threadIdx.x;
  if (idx >= ntot) return;
  hw[idx] = (hw[idx] + fw[idx]) * 0.5f;
}

// -------------------------------------------------------------------------
// Fused MLP: logits = relu(relu(hA@W1+b1)@W2+b2)@W3+b3.
// One block per 16 rows; hA tile + both intermediates live in LDS.
// Stage1 N=256 over 8 waves; stage2 N=32 over 2 waves; stage3 N=10 by wave 0.
// -------------------------------------------------------------------------
__global__ __launch_bounds__(256) void mlp_fused(
    const float* __restrict__ hA,
    const float* __restrict__ W1, const float* __restrict__ b1,
    const float* __restrict__ W2, const float* __restrict__ b2,
    const float* __restrict__ W3, const float* __restrict__ b3,
    float* __restrict__ logits) {
  __shared__ __align__(16) float hAs[16 * AS_STRIDE];  // 16 x 128 (padded)
  __shared__ __align__(16) float x1s[16 * X1_STRIDE];  // 16 x 256 (padded)
  __shared__ __align__(16) float x2s[16 * X2_STRIDE];  // 16 x 32  (padded)
  const int lane = threadIdx.x & 31;
  const int wv = threadIdx.x >> 5;
  const int m0 = blockIdx.x * 16;
  const int l15 = lane & 15;
  const int koff = (lane >> 4) * 2;
  const int rhalf = (lane >> 4) * 8;

  // stage hA tile: 16 x 128 = 512 float4
  for (int i = threadIdx.x; i < 512; i += 256) {
    int r = i >> 5;
    int c = (i & 31) << 2;
    float4 v = *(const float4*)(hA + (size_t)(m0 + r) * HH + c);
    *(float4*)(hAs + r * AS_STRIDE + c) = v;
  }
  __syncthreads();

  // ---- stage 1: x1 = relu(hA[16x128] @ W1[128x256] + b1) ----
  const float* ap0 = hAs + l15 * AS_STRIDE + koff;
#pragma unroll
  for (int t = 0; t < 2; ++t) {
    const int n0 = (wv * 2 + t) * 16;
    v8f acc = {};
    for (int k = 0; k < HH; k += 4) {
      v2f a, b;
      a.x = ap0[k];
      a.y = ap0[k + 1];
      const float* bp = W1 + (size_t)(k + koff) * LH + n0 + l15;
      b.x = bp[0];
      b.y = bp[LH];
      acc = __builtin_amdgcn_wmma_f32_16x16x4_f32(false, a, false, b,
                                                  (short)0, acc, false, false);
    }
    float bb = b1[n0 + l15];
#pragma unroll
    for (int v = 0; v < 8; ++v)
      x1s[(rhalf + v) * X1_STRIDE + n0 + l15] = fmaxf(acc[v] + bb, 0.0f);
  }
  __syncthreads();

  // ---- stage 2: x2 = relu(x1[16x256] @ W2[256x32] + b2) ----
  if (wv < 2) {
    const int n0 = wv * 16;
    const float* ap1 = x1s + l15 * X1_STRIDE + koff;
    v8f acc = {};
    for (int k = 0; k < LH; k += 4) {
      v2f a, b;
      a.x = ap1[k];
      a.y = ap1[k + 1];
      const float* bp = W2 + (size_t)(k + koff) * 32 + n0 + l15;
      b.x = bp[0];
      b.y = bp[32];
      acc = __builtin_amdgcn_wmma_f32_16x16x4_f32(false, a, false, b,
                                                  (short)0, acc, false, false);
    }
    float bb = b2[n0 + l15];
#pragma unroll
    for (int v = 0; v < 8; ++v)
      x2s[(rhalf + v) * X2_STRIDE + n0 + l15] = fmaxf(acc[v] + bb, 0.0f);
  }
  __syncthreads();

  // ---- stage 3: logits = x2[16x32] @ W3[32x10] + b3 (cols 0..9) ----
  if (wv == 0) {
    const bool colok = (l15 < OUTC);
    const int colc = colok ? l15 : (OUTC - 1);
    const float* ap2 = x2s + l15 * X2_STRIDE + koff;
    v8f acc = {};
    for (int k = 0; k < 32; k += 4) {
      v2f a, b;
      a.x = ap2[k];
      a.y = ap2[k + 1];
      float bx = W3[(size_t)(k + koff) * OUTC + colc];
      float by = W3[(size_t)(k + koff + 1) * OUTC + colc];
      b.x = colok ? bx : 0.0f;
      b.y = colok ? by : 0.0f;
      acc = __builtin_amdgcn_wmma_f32_16x16x4_f32(false, a, false, b,
                                                  (short)0, acc, false, false);
    }
    if (colok) {
      float bb = b3[l15];
#pragma unroll
      for (int v = 0; v < 8; ++v)
        logits[(size_t)(m0 + rhalf + v) * OUTC + l15] = acc[v] + bb;
    }
  }
}

// ---------------------------------------------------------------------------
extern "C" void kernel_launch(void* const* d_in, const int* in_sizes, int n_in,
                              void* d_out, int out_size, void* d_ws, size_t ws_size,
                              hipStream_t stream) {
  (void)in_sizes; (void)n_in; (void)out_size; (void)ws_size;
  const float* feat_a      = (const float*)d_in[0];
  const float* feat_w      = (const float*)d_in[1];
  const int*   src_tic     = (const int*)d_in[2];
  const int*   dst_tic     = (const int*)d_in[3];
  const float* w_tic       = (const float*)d_in[4];
  const int*   src_rel     = (const int*)d_in[5];
  const int*   dst_rel     = (const int*)d_in[6];
  const float* w_rel       = (const float*)d_in[7];
  const float* W_self_tic  = (const float*)d_in[8];
  const float* W_neigh_tic = (const float*)d_in[9];
  const float* b_tic       = (const float*)d_in[10];
  const float* W_self_rel  = (const float*)d_in[11];
  const float* W_neigh_rel = (const float*)d_in[12];
  const float* b_rel       = (const float*)d_in[13];
  const float* W1          = (const float*)d_in[14];
  const float* b1          = (const float*)d_in[15];
  const float* W2          = (const float*)d_in[16];
  const float* b2          = (const float*)d_in[17];
  const float* W3          = (const float*)d_in[18];
  const float* b3          = (const float*)d_in[19];

  // workspace: agg_tic[NA*128], cnt_tic[NA], agg_rel[NW*128], cnt_rel[NW]
  float* agg_tic = (float*)d_ws;
  float* cnt_tic = agg_tic + (size_t)NA * DD;
  float* agg_rel = cnt_tic + NA;
  float* cnt_rel = agg_rel + (size_t)NW * DD;
  size_t zero_bytes =
      ((size_t)NA * DD + NA + (size_t)NW * DD + NW) * sizeof(float);
  hipMemsetAsync(d_ws, 0, zero_bytes, stream);

  // outputs: logits | h_acoustic | h_word  (flat, in return order)
  float* logits = (float*)d_out;
  float* h_ac   = logits + (size_t)NA * OUTC;
  float* h_word = h_ac + (size_t)NA * HH;

  // edge aggregation (segment sum + count)
  scatter_edges<<<(E1 * 32) / 256, 256, 0, stream>>>(
      feat_a, src_tic, dst_tic, w_tic, agg_tic, cnt_tic, E1);
  scatter_edges<<<(E2 * 32) / 256, 256, 0, stream>>>(
      feat_a, src_rel, dst_rel, w_rel, agg_rel, cnt_rel, E2);
  seg_div<<<(NA * DD) / 256, 256, 0, stream>>>(agg_tic, cnt_tic, NA * DD);
  seg_div<<<(NW * DD) / 256, 256, 0, stream>>>(agg_rel, cnt_rel, NW * DD);

  // h_acoustic = feat_a@W_self_tic + agg_tic@W_neigh_tic + b_tic
  gemm_dual_128<<<NA / 16, 256, 0, stream>>>(feat_a, W_self_tic, agg_tic,
                                             W_neigh_tic, b_tic, h_ac);
  // h_rel into h_word slot, then combine with identity half
  gemm_dual_128<<<NW / 16, 256, 0, stream>>>(feat_w, W_self_rel, agg_rel,
                                             W_neigh_rel, b_rel, h_word);
  word_combine<<<(NW * DD) / 256, 256, 0, stream>>>(h_word, feat_w, NW * DD);

  // fused 3-layer MLP -> logits
  mlp_fused<<<NA / 16, 256, 0, stream>>>(h_ac, W1, b1, W2, b2, W3, b3, logits);
}